// GCN_34540126994447
// MI455X (gfx1250) — compile-verified
//
#include <hip/hip_runtime.h>

typedef __attribute__((ext_vector_type(2))) float v2f;
typedef __attribute__((ext_vector_type(8))) float v8f;

#define NNODES 50000
#define DHID   128
#define NCLS   10

// ---------------------------------------------------------------------------
// Normalization kernels
// ---------------------------------------------------------------------------
__global__ void deg_init_kernel(float* deg, int n) {
    int i = blockIdx.x * blockDim.x + threadIdx.x;
    if (i < n) deg[i] = 1.0f;                 // self-loop weight
}

__global__ void deg_edges_kernel(const int* __restrict__ dst,
                                 const float* __restrict__ w,
                                 float* deg, int nE) {
    int e = blockIdx.x * blockDim.x + threadIdx.x;
    if (e < nE) atomicAdd(&deg[dst[e]], w[e]);
}

__global__ void dinv_kernel(const float* __restrict__ deg, float* dinv, int n) {
    int i = blockIdx.x * blockDim.x + threadIdx.x;
    if (i < n) {
        float d = deg[i];                      // >= 1 always (self loop)
        dinv[i] = __frsqrt_rn(d);
    }
}

__global__ void norm_kernel(const int* __restrict__ src,
                            const int* __restrict__ dst,
                            const float* __restrict__ w,
                            const float* __restrict__ dinv,
                            float* norm, int nE) {
    int e = blockIdx.x * blockDim.x + threadIdx.x;
    if (e < nE) norm[e] = dinv[src[e]] * w[e] * dinv[dst[e]];
}

// ---------------------------------------------------------------------------
// GEMM: H[M,128] = X[M,K] @ W[K,128]  using V_WMMA_F32_16X16X4_F32
//   block = 256 threads (8 waves), computes a 128x128 tile of H.
//   K staged through LDS in chunks of 32 (zero padded at the tail).
//   W chunk stored K-pair interleaved (float2) so each B fragment is one
//   contiguous 8-byte ds_load straight into an aligned VGPR pair.
// ---------------------------------------------------------------------------
#define XS_STRIDE 34    // floats per row: 32 k + pad (conflict-free A b64 reads)
#define WT_STRIDE 144   // float2 per kpair row: 128 cols + pad (stride 288 dw,
                        // = 32 mod 64 banks -> lane halves in disjoint windows)

__global__ __launch_bounds__(256) void gemm128_wmma_kernel(
    const float* __restrict__ X, const float* __restrict__ W,
    float* __restrict__ H, int M, int K)
{
    __shared__ float  Xs[128 * XS_STRIDE];
    __shared__ float2 Wt[16 * WT_STRIDE];

    const int tid  = threadIdx.x;
    const int lane = tid & 31;
    const int wave = tid >> 5;
    const int l16  = lane & 15;
    const bool hi  = (lane >= 16);
    const int rowBlock = blockIdx.x * 128;
    const int rowLocal = wave * 16 + l16;     // A-frag row for this lane

    v8f acc[8];
    #pragma unroll
    for (int t = 0; t < 8; ++t)
        #pragma unroll
        for (int r = 0; r < 8; ++r) acc[t][r] = 0.0f;

    const int nch = (K + 31) / 32;
    for (int ch = 0; ch < nch; ++ch) {
        const int k0 = ch * 32;

        // ---- stage X chunk: 128 rows x 32 k  (1024 float4, 4 per thread)
        #pragma unroll
        for (int j = 0; j < 4; ++j) {
            int f  = tid * 4 + j;             // 0..1023
            int r  = f >> 3;                  // row 0..127
            int kb = (f & 7) * 4;             // k offset 0,4,...,28
            int gr = rowBlock + r;
            float4 v = make_float4(0.f, 0.f, 0.f, 0.f);
            if (gr < M) {
                int kg = k0 + kb;
                if (kg + 3 < K) {
                    v = *(const float4*)(X + (size_t)gr * K + kg);
                } else {
                    float t0 = (kg + 0 < K) ? X[(size_t)gr * K + kg + 0] : 0.f;
                    float t1 = (kg + 1 < K) ? X[(size_t)gr * K + kg + 1] : 0.f;
                    float t2 = (kg + 2 < K) ? X[(size_t)gr * K + kg + 2] : 0.f;
                    float t3 = (kg + 3 < K) ? X[(size_t)gr * K + kg + 3] : 0.f;
                    v = make_float4(t0, t1, t2, t3);
                }
            }
            float* dp = &Xs[r * XS_STRIDE + kb];
            ((float2*)dp)[0] = make_float2(v.x, v.y);
            ((float2*)dp)[1] = make_float2(v.z, v.w);
        }

        // ---- stage W chunk interleaved: Wt[kp][col] = {W[2kp][col], W[2kp+1][col]}
        //      512 tasks (16 kpairs x 32 col-quads), 2 per thread
        #pragma unroll
        for (int j = 0; j < 2; ++j) {
            int f  = tid * 2 + j;             // 0..511
            int kp = f >> 5;                  // kpair 0..15
            int cb = (f & 31) * 4;            // col 0,4,...,124
            int ka = k0 + 2 * kp;
            float4 va = make_float4(0.f, 0.f, 0.f, 0.f);
            float4 vb = make_float4(0.f, 0.f, 0.f, 0.f);
            if (ka < K)     va = *(const float4*)(W + (size_t)ka * DHID + cb);
            if (ka + 1 < K) vb = *(const float4*)(W + (size_t)(ka + 1) * DHID + cb);
            float4 w0 = make_float4(va.x, vb.x, va.y, vb.y);
            float4 w1 = make_float4(va.z, vb.z, va.w, vb.w);
            *(float4*)&Wt[kp * WT_STRIDE + cb]     = w0;
            *(float4*)&Wt[kp * WT_STRIDE + cb + 2] = w1;
        }

        __syncthreads();

        // ---- 8 K-substeps of 4, 8 N-tiles of 16 -> 64 WMMA per wave per chunk
        #pragma unroll
        for (int s = 0; s < 8; ++s) {
            const int kp = s * 2 + (hi ? 1 : 0);   // this lane's K-pair index
            // A 16x4 frag: one b64 read (two consecutive K values of this row)
            v2f a = *(const v2f*)&Xs[rowLocal * XS_STRIDE + kp * 2];
            // B frags: one b64 read each, already K-pair interleaved
            float2 bf[8];
            #pragma unroll
            for (int nt = 0; nt < 8; ++nt)
                bf[nt] = Wt[kp * WT_STRIDE + nt * 16 + l16];
            #pragma unroll
            for (int nt = 0; nt < 8; ++nt) {
                v2f b;
                b.x = bf[nt].x;
                b.y = bf[nt].y;
                acc[nt] = __builtin_amdgcn_wmma_f32_16x16x4_f32(
                    /*neg_a=*/false, a, /*neg_b=*/false, b,
                    /*c_mod=*/(short)0, acc[nt],
                    /*reuse_a=*/false, /*reuse_b=*/false);
            }
        }
        __syncthreads();
    }

    // ---- store: VGPR r -> row (wave*16 + r + hi*8), col nt*16 + l16
    #pragma unroll
    for (int nt = 0; nt < 8; ++nt) {
        const int col = nt * 16 + l16;
        #pragma unroll
        for (int r = 0; r < 8; ++r) {
            int m = rowBlock + wave * 16 + r + (hi ? 8 : 0);
            if (m < M) H[(size_t)m * DHID + col] = acc[nt][r];
        }
    }
}

// ---------------------------------------------------------------------------
// Aggregation: out = segment_sum(h[src]*norm) + h*dinv^2 (self loop)
// ---------------------------------------------------------------------------
__global__ void agg_self_kernel(const float* __restrict__ H,
                                const float* __restrict__ dinv,
                                float* __restrict__ OUT, int n) {
    int i = blockIdx.x * blockDim.x + threadIdx.x;
    if (i < n * DHID) {
        float s = dinv[i >> 7];
        OUT[i] = H[i] * s * s;
    }
}

__global__ __launch_bounds__(256) void agg_edges_kernel(
    const float* __restrict__ H,
    const int* __restrict__ src, const int* __restrict__ dst,
    const float* __restrict__ norm,
    float* __restrict__ OUT, int nE)
{
    int e = blockIdx.x * 2 + (threadIdx.x >> 7);
    int d = threadIdx.x & 127;
    if (e < nE) {
        int s = src[e];
        int t = dst[e];
        float v = H[(size_t)s * DHID + d] * norm[e];
        atomicAdd(&OUT[(size_t)t * DHID + d], v);
    }
}

__global__ void bias_relu_kernel(float* __restrict__ OUT,
                                 const float* __restrict__ b, int n) {
    int i = blockIdx.x * blockDim.x + threadIdx.x;
    if (i < n * DHID) {
        float v = OUT[i] + b[i & (DHID - 1)];
        OUT[i] = v > 0.f ? v : 0.f;
    }
}

// ---------------------------------------------------------------------------
// Classifier: logits[N,10] = H[N,128] @ Wc[128,10] + bc  (tiny, VALU)
// ---------------------------------------------------------------------------
__global__ void classifier_kernel(const float* __restrict__ H,
                                  const float* __restrict__ Wc,
                                  const float* __restrict__ bc,
                                  float* __restrict__ out, int n) {
    int i = blockIdx.x * blockDim.x + threadIdx.x;
    if (i < n * NCLS) {
        int row = i / NCLS;
        int c   = i - row * NCLS;
        float s = bc[c];
        const float* h = H + (size_t)row * DHID;
        #pragma unroll 8
        for (int k = 0; k < DHID; ++k) s += h[k] * Wc[k * NCLS + c];
        out[i] = s;
    }
}

// ---------------------------------------------------------------------------
// Launch
// ---------------------------------------------------------------------------
extern "C" void kernel_launch(void* const* d_in, const int* in_sizes, int n_in,
                              void* d_out, int out_size, void* d_ws, size_t ws_size,
                              hipStream_t stream) {
    const float* x  = (const float*)d_in[0];
    const int*   ei = (const int*)d_in[1];
    const float* ew = (const float*)d_in[2];
    const float* W1 = (const float*)d_in[3];
    const float* b1 = (const float*)d_in[4];
    const float* W2 = (const float*)d_in[5];
    const float* b2 = (const float*)d_in[6];
    const float* Wc = (const float*)d_in[7];
    const float* bc = (const float*)d_in[8];
    float* out = (float*)d_out;

    const int n  = NNODES;
    const int E  = in_sizes[1] / 2;
    const int K1 = in_sizes[0] / n;            // 500
    const int* src = ei;
    const int* dst = ei + E;

    // workspace layout (256B aligned)
    char* ws = (char*)d_ws;
    size_t off = 0;
    auto alloc = [&](size_t bytes) { char* p = ws + off; off = (off + bytes + 255) & ~(size_t)255; return p; };
    float* deg  = (float*)alloc((size_t)n * 4);
    float* dinv = (float*)alloc((size_t)n * 4);
    float* norm = (float*)alloc((size_t)E * 4);
    float* bufA = (float*)alloc((size_t)n * DHID * 4);
    float* bufB = (float*)alloc((size_t)n * DHID * 4);
    (void)ws_size;

    const int T = 256;
    // ---- gcn_norm
    deg_init_kernel <<<(n + T - 1) / T, T, 0, stream>>>(deg, n);
    deg_edges_kernel<<<(E + T - 1) / T, T, 0, stream>>>(dst, ew, deg, E);
    dinv_kernel     <<<(n + T - 1) / T, T, 0, stream>>>(deg, dinv, n);
    norm_kernel     <<<(E + T - 1) / T, T, 0, stream>>>(src, dst, ew, dinv, norm, E);

    const int gemmBlocks = (n + 127) / 128;
    const int elemBlocks = (n * DHID + T - 1) / T;
    const int edgeBlocks = (E + 1) / 2;

    // ---- layer 1: h = relu(agg(x @ W1) + b1)
    gemm128_wmma_kernel<<<gemmBlocks, T, 0, stream>>>(x, W1, bufA, n, K1);
    agg_self_kernel    <<<elemBlocks, T, 0, stream>>>(bufA, dinv, bufB, n);
    agg_edges_kernel   <<<edgeBlocks, T, 0, stream>>>(bufA, src, dst, norm, bufB, E);
    bias_relu_kernel   <<<elemBlocks, T, 0, stream>>>(bufB, b1, n);

    // ---- layer 2: h = relu(agg(h @ W2) + b2)
    gemm128_wmma_kernel<<<gemmBlocks, T, 0, stream>>>(bufB, W2, bufA, n, DHID);
    agg_self_kernel    <<<elemBlocks, T, 0, stream>>>(bufA, dinv, bufB, n);
    agg_edges_kernel   <<<edgeBlocks, T, 0, stream>>>(bufA, src, dst, norm, bufB, E);
    bias_relu_kernel   <<<elemBlocks, T, 0, stream>>>(bufB, b2, n);

    // ---- classifier
    classifier_kernel<<<(n * NCLS + T - 1) / T, T, 0, stream>>>(bufB, Wc, bc, out, n);
}